// AnomalyAttention_33414845563379
// MI455X (gfx1250) — compile-verified
//
#include <hip/hip_runtime.h>
#include <hip/hip_bf16.h>

// AnomalyAttention for MI455X (gfx1250, wave32).
// B=1, L=8192, D=64. Outputs concatenated: prior[L*L], series[L*L], out[L*D].
// Write-bandwidth bound (~512MB of L*L f32 output). Matrix math with
// full-precision V_WMMA_F32_16X16X4_F32; K/V tiles double-buffered in LDS via
// gfx1250 async-to-LDS copies (ASYNCcnt).

typedef __attribute__((ext_vector_type(2))) float v2f;
typedef __attribute__((ext_vector_type(8))) float v8f;

constexpr int LQ = 8192;
constexpr int DK = 64;
constexpr int NT = LQ / 16;          // 512 j-tiles
constexpr int KST = 68;              // padded LDS row stride (floats)
constexpr float INV_SQRT_2PI = 0.39894228040143267794f;

// ---- gfx1250 async copy helpers (ASYNCcnt) --------------------------------
__device__ __forceinline__ void async_copy_b128(const void* lds_dst, const void* gsrc) {
  // per-lane 16B global->LDS; VDST = LDS byte offset (low 32 bits of generic
  // LDS address == AS(3) offset), VADDR = 64-bit global address.
  asm volatile("global_load_async_to_lds_b128 %0, %1, off"
               :: "v"((unsigned)(uintptr_t)lds_dst), "v"(gsrc)
               : "memory");
}
__device__ __forceinline__ void wait_async0() {
  asm volatile("s_wait_asynccnt 0" ::: "memory");
}

// ---------------------------------------------------------------------------
// Kernel A1: partial[c][i] = sum_{j in chunk c} exp(-(i-j)^2/(2 sigma_i^2))
// ---------------------------------------------------------------------------
__global__ __launch_bounds__(256) void prior_partial_k(const float* __restrict__ Sigma,
                                                       float* __restrict__ partial) {
  const int i = blockIdx.x * blockDim.x + threadIdx.x;
  const int c = blockIdx.y;                       // 8 chunks of 1024 columns
  const float sg = Sigma[i];
  const float a  = -0.5f * __builtin_amdgcn_rcpf(sg * sg);
  float sum = 0.0f;
  const int j0 = c * (LQ / 8);
#pragma unroll 4
  for (int j = j0; j < j0 + LQ / 8; ++j) {
    const float d = (float)(i - j);
    sum += __expf(a * d * d);
  }
  partial[(size_t)c * LQ + i] = sum;
}

// Kernel A2: rs[i] = inv_norm_i * sum_c partial[c][i]  (fixed order: determin.)
__global__ __launch_bounds__(256) void prior_reduce_k(const float* __restrict__ Sigma,
                                                      const float* __restrict__ partial,
                                                      float* __restrict__ rs) {
  const int i = blockIdx.x * blockDim.x + threadIdx.x;
  float s = 0.0f;
#pragma unroll
  for (int c = 0; c < 8; ++c) s += partial[(size_t)c * LQ + i];
  const float sg = Sigma[i];
  rs[i] = INV_SQRT_2PI * __builtin_amdgcn_rcpf(sg) * s;
}

// ---------------------------------------------------------------------------
// Kernel B: prior[i][j] = inv_i * exp(a_i*(i-j)^2) / rs[j]   (float4 stores)
// ---------------------------------------------------------------------------
__global__ __launch_bounds__(256) void prior_write_k(const float* __restrict__ Sigma,
                                                     const float* __restrict__ rs,
                                                     float* __restrict__ prior) {
  const int i  = blockIdx.y;
  const int j4 = blockIdx.x * blockDim.x + threadIdx.x;
  const int j  = j4 * 4;
  const float sg  = Sigma[i];
  const float inv = INV_SQRT_2PI * __builtin_amdgcn_rcpf(sg);
  const float a   = -0.5f * __builtin_amdgcn_rcpf(sg * sg);
  const float4 r  = ((const float4*)rs)[j4];
  const float d0 = (float)(i - j),     d1 = (float)(i - j - 1);
  const float d2 = (float)(i - j - 2), d3 = (float)(i - j - 3);
  float4 o;
  o.x = inv * __expf(a * d0 * d0) * __builtin_amdgcn_rcpf(r.x);
  o.y = inv * __expf(a * d1 * d1) * __builtin_amdgcn_rcpf(r.y);
  o.z = inv * __expf(a * d2 * d2) * __builtin_amdgcn_rcpf(r.z);
  o.w = inv * __expf(a * d3 * d3) * __builtin_amdgcn_rcpf(r.w);
  ((float4*)prior)[(size_t)i * (LQ / 4) + j4] = o;
}

// ---------------------------------------------------------------------------
// Kernel C: softmax stats. Wave = 16-row strip (block = 128 rows). K tiles
// double-buffered in LDS via async copies. Per-lane online (m,l) over this
// lane's column subset; one shuffle-merge at the end.
// ---------------------------------------------------------------------------
__global__ __launch_bounds__(256) void stats_k(const float* __restrict__ Q,
                                               const float* __restrict__ Km,
                                               const int* __restrict__ dkp,
                                               float* __restrict__ ws_m,
                                               float* __restrict__ ws_l) {
  __shared__ __align__(16) float ktile[2][16 * KST];

  const int lane = threadIdx.x & 31;
  const int wave = threadIdx.x >> 5;
  const int lr   = lane & 15;
  const int h    = lane >> 4;
  const int i0   = blockIdx.x * 128 + wave * 16;
  const float scale = rsqrtf((float)dkp[0]);

  // cooperative tile-fill mapping: 256 slots = 16 rows x 16 groups of 4 floats
  const int srow = threadIdx.x >> 4;
  const int scol = (threadIdx.x & 15) * 4;

  v2f af[16];
#pragma unroll
  for (int k = 0; k < 16; ++k)
    af[k] = *(const v2f*)(Q + (size_t)(i0 + lr) * DK + 4 * k + 2 * h);

  float mrun[8], lrun[8];
#pragma unroll
  for (int v = 0; v < 8; ++v) { mrun[v] = -1e30f; lrun[v] = 0.0f; }

  // prologue: stage tile 0
  async_copy_b128(&ktile[0][srow * KST + scol], Km + (size_t)srow * DK + scol);
  wait_async0();
  __syncthreads();

  for (int jt = 0; jt < NT; ++jt) {
    const int cur = jt & 1;
    if (jt + 1 < NT)
      async_copy_b128(&ktile[cur ^ 1][srow * KST + scol],
                      Km + (size_t)((jt + 1) * 16 + srow) * DK + scol);

    const float* kb = &ktile[cur][(size_t)lr * KST + 2 * h];
    v8f c0 = {}, c1 = {};
#pragma unroll
    for (int k = 0; k < 8; ++k) {
      v2f b = *(const v2f*)(kb + 4 * k);
      c0 = __builtin_amdgcn_wmma_f32_16x16x4_f32(false, af[k], false, b,
                                                 (short)0, c0, false, false);
    }
#pragma unroll
    for (int k = 8; k < 16; ++k) {
      v2f b = *(const v2f*)(kb + 4 * k);
      c1 = __builtin_amdgcn_wmma_f32_16x16x4_f32(false, af[k], false, b,
                                                 (short)0, c1, false, false);
    }
#pragma unroll
    for (int v = 0; v < 8; ++v) {
      const float s  = (c0[v] + c1[v]) * scale;
      const float mn = fmaxf(mrun[v], s);
      lrun[v] = lrun[v] * __expf(mrun[v] - mn) + __expf(s - mn);
      mrun[v] = mn;
    }
    wait_async0();
    __syncthreads();
  }

  // merge the 16 lanes of each half, write stats
#pragma unroll
  for (int v = 0; v < 8; ++v) {
    float m = mrun[v], l = lrun[v];
#pragma unroll
    for (int mask = 1; mask <= 8; mask <<= 1) {
      const float om = __shfl_xor(m, mask);
      const float ol = __shfl_xor(l, mask);
      const float mn = fmaxf(m, om);
      l = l * __expf(m - mn) + ol * __expf(om - mn);
      m = mn;
    }
    if (lr == 0) {
      const int row = i0 + v + 8 * h;
      ws_m[row] = m;
      ws_l[row] = l;
    }
  }
}

// ---------------------------------------------------------------------------
// Kernel D: recompute logits from LDS-staged K, write series = exp(s-m)/l,
// accumulate out = S@V with V also LDS-staged; P transposed via padded LDS.
// ---------------------------------------------------------------------------
__global__ __launch_bounds__(256) void attn_k(const float* __restrict__ Q,
                                              const float* __restrict__ Km,
                                              const float* __restrict__ V,
                                              const int* __restrict__ dkp,
                                              const float* __restrict__ ws_m,
                                              const float* __restrict__ ws_l,
                                              float* __restrict__ series,
                                              float* __restrict__ outp) {
  __shared__ __align__(16) float ktile[2][16 * KST];
  __shared__ __align__(16) float vtile[2][16 * KST];
  __shared__ __align__(16) float plds[8][16][17];

  const int lane = threadIdx.x & 31;
  const int wave = threadIdx.x >> 5;
  const int lr   = lane & 15;
  const int h    = lane >> 4;
  const int i0   = blockIdx.x * 128 + wave * 16;
  const float scale = rsqrtf((float)dkp[0]);

  const int srow = threadIdx.x >> 4;
  const int scol = (threadIdx.x & 15) * 4;

  v2f af[16];
#pragma unroll
  for (int k = 0; k < 16; ++k)
    af[k] = *(const v2f*)(Q + (size_t)(i0 + lr) * DK + 4 * k + 2 * h);

  float mv[8], linv[8];
#pragma unroll
  for (int v = 0; v < 8; ++v) {
    const int row = i0 + v + 8 * h;
    mv[v]   = ws_m[row];
    linv[v] = __builtin_amdgcn_rcpf(ws_l[row]);
  }

  v8f oc[4] = {{}, {}, {}, {}};        // 16 rows x 64 cols of `out`

  async_copy_b128(&ktile[0][srow * KST + scol], Km + (size_t)srow * DK + scol);
  async_copy_b128(&vtile[0][srow * KST + scol], V  + (size_t)srow * DK + scol);
  wait_async0();
  __syncthreads();

  for (int jt = 0; jt < NT; ++jt) {
    const int cur = jt & 1;
    const int j0  = jt * 16;
    if (jt + 1 < NT) {
      const size_t nrow = (size_t)(j0 + 16 + srow) * DK + scol;
      async_copy_b128(&ktile[cur ^ 1][srow * KST + scol], Km + nrow);
      async_copy_b128(&vtile[cur ^ 1][srow * KST + scol], V + nrow);
    }

    const float* kb = &ktile[cur][(size_t)lr * KST + 2 * h];
    v8f c0 = {}, c1 = {};
#pragma unroll
    for (int k = 0; k < 8; ++k) {
      v2f b = *(const v2f*)(kb + 4 * k);
      c0 = __builtin_amdgcn_wmma_f32_16x16x4_f32(false, af[k], false, b,
                                                 (short)0, c0, false, false);
    }
#pragma unroll
    for (int k = 8; k < 16; ++k) {
      v2f b = *(const v2f*)(kb + 4 * k);
      c1 = __builtin_amdgcn_wmma_f32_16x16x4_f32(false, af[k], false, b,
                                                 (short)0, c1, false, false);
    }

    float p[8];
#pragma unroll
    for (int v = 0; v < 8; ++v)
      p[v] = __expf((c0[v] + c1[v]) * scale - mv[v]) * linv[v];

    // write series tile (lanes 0-15 cover a row contiguously)
#pragma unroll
    for (int v = 0; v < 8; ++v)
      series[(size_t)(i0 + v + 8 * h) * LQ + j0 + lr] = p[v];

    // per-wave C-layout -> A-layout transpose through padded LDS
#pragma unroll
    for (int v = 0; v < 8; ++v)
      plds[wave][v + 8 * h][lr] = p[v];
    __builtin_amdgcn_wave_barrier();   // LDS ops are in-order per wave

    v2f pa[4];
#pragma unroll
    for (int kk = 0; kk < 4; ++kk) {
      pa[kk].x = plds[wave][lr][4 * kk + 2 * h];
      pa[kk].y = plds[wave][lr][4 * kk + 1 + 2 * h];
    }

    // out(16x64) += P(16x16) * V(16x64): V B-frags from LDS
#pragma unroll
    for (int kk = 0; kk < 4; ++kk) {
#pragma unroll
      for (int dd = 0; dd < 4; ++dd) {
        v2f vb;
        vb.x = vtile[cur][(4 * kk + 2 * h) * KST + 16 * dd + lr];
        vb.y = vtile[cur][(4 * kk + 2 * h + 1) * KST + 16 * dd + lr];
        oc[dd] = __builtin_amdgcn_wmma_f32_16x16x4_f32(false, pa[kk], false, vb,
                                                       (short)0, oc[dd], false, false);
      }
    }

    wait_async0();
    __syncthreads();
  }

#pragma unroll
  for (int dd = 0; dd < 4; ++dd)
#pragma unroll
    for (int v = 0; v < 8; ++v)
      outp[(size_t)(i0 + v + 8 * h) * DK + 16 * dd + lr] = oc[dd][v];
}

// ---------------------------------------------------------------------------
extern "C" void kernel_launch(void* const* d_in, const int* in_sizes, int n_in,
                              void* d_out, int out_size, void* d_ws, size_t ws_size,
                              hipStream_t stream) {
  const float* Sigma = (const float*)d_in[0];
  const float* Q     = (const float*)d_in[1];
  const float* K     = (const float*)d_in[2];
  const float* V     = (const float*)d_in[3];
  const int*   dk    = (const int*)d_in[4];

  float* out     = (float*)d_out;
  float* prior   = out;                                  // L*L
  float* series  = out + (size_t)LQ * LQ;                // L*L
  float* attnout = out + 2 * (size_t)LQ * LQ;            // L*D

  float* ws      = (float*)d_ws;
  float* rs      = ws;             // L floats: prior column divisors
  float* wsm     = ws + LQ;        // L floats: softmax row max
  float* wsl     = ws + 2 * LQ;    // L floats: softmax row denom
  float* partial = ws + 3 * LQ;    // 8*L floats: prior partial sums

  prior_partial_k<<<dim3(LQ / 256, 8), 256, 0, stream>>>(Sigma, partial);
  stats_k<<<LQ / 128, 256, 0, stream>>>(Q, K, dk, wsm, wsl);
  prior_reduce_k<<<LQ / 256, 256, 0, stream>>>(Sigma, partial, rs);
  prior_write_k<<<dim3(LQ / 1024, LQ), 256, 0, stream>>>(Sigma, rs, prior);
  attn_k<<<LQ / 128, 256, 0, stream>>>(Q, K, V, dk, wsm, wsl, series, attnout);
}